// ScaleSpaceDetector_72610717106657
// MI455X (gfx1250) — compile-verified
//
#include <hip/hip_runtime.h>
#include <hip/hip_bf16.h>

typedef float v2f __attribute__((ext_vector_type(2)));
typedef float v8f __attribute__((ext_vector_type(8)));
typedef unsigned long long u64;

#define BATCH     4
#define NUM_FEATS 2000
#define MR_SIZE   5.192f
#define TEMP_F    5.0f
// candidate counts: 512^2 + 256^2 + 128^2 + 64^2 + 32^2
#define NCAND     349184
#define CHUNK     4096
#define KEEP      2048

// sigma_i = 1.6 * 2^(i/3); radius = round(3*sigma)
__constant__ float d_sig[3] = {1.6f, 2.0158736798317971f, 2.5398416831491192f};

__device__ __forceinline__ int iclamp(int v, int lo, int hi) {
  return v < lo ? lo : (v > hi ? hi : v);
}

// ---------------------------------------------------------------------------
// Downsample base[:, ::2, ::2]
// ---------------------------------------------------------------------------
__global__ void k_downsample(const float* __restrict__ in, float* __restrict__ out,
                             int H, int W) {
  const int Ho = H >> 1, Wo = W >> 1;
  int idx = blockIdx.x * blockDim.x + threadIdx.x;
  int total = BATCH * Ho * Wo;
  if (idx >= total) return;
  int b = idx / (Ho * Wo);
  int p = idx - b * (Ho * Wo);
  int y = p / Wo, x = p - y * Wo;
  out[idx] = in[((size_t)b * H + 2 * y) * W + 2 * x];
}

// ---------------------------------------------------------------------------
// Build normalized, zero-padded Gaussian band into LDS: gz[16+d] = g[d]/sum
// for 0 <= d <= 2R, zero elsewhere (indices 0..63).
// ---------------------------------------------------------------------------
template <int R>
__device__ __forceinline__ void build_band(float* gz, int ft, float sig) {
  if (ft < 64) {
    float ssum = 0.f;
#pragma unroll
    for (int t = 0; t <= 2 * R; ++t) {
      float xo = (float)(t - R);
      ssum += expf(-0.5f * (xo / sig) * (xo / sig));
    }
    int d = ft - 16;
    float v = 0.f;
    if (d >= 0 && d <= 2 * R) {
      float xo = (float)(d - R);
      v = expf(-0.5f * (xo / sig) * (xo / sig)) / ssum;
    }
    gz[ft] = v;
  }
}

// ---------------------------------------------------------------------------
// Horizontal blur: O(16x16) = A(16xK) * G(Kx16), K = 16+2R,
//   A[m,k] = In[y0+m, x0-R+k] (zero-padded, staged in LDS), G[k,n] = g[k-n].
// Block = 4 waves, one 16x16 tile per wave; fully unrolled WMMA chain.
// ---------------------------------------------------------------------------
template <int R>
__global__ void k_hblur_t(const float* __restrict__ in, float* __restrict__ out,
                          int H, int W, float sig) {
  constexpr int K   = 16 + 2 * R;
  constexpr int NCH = (K + 3) / 4;
  constexpr int KC  = NCH * 4;   // padded K (multiple of 4)
  constexpr int KP  = 34;        // LDS row stride (even, bank-spreading)
  __shared__ float gz[64];
  __shared__ __align__(16) float tile[4][16 * KP];

  const int lane = threadIdx.x;            // 0..31
  const int wv   = threadIdx.y;            // 0..3
  const int ft   = wv * 32 + lane;

  build_band<R>(gz, ft, sig);

  const int b  = blockIdx.z;
  const int x0 = blockIdx.x << 4;
  const int y0 = ((blockIdx.y << 2) + wv) << 4;
  const float* pl = in + (size_t)b * H * W;

  // stage zero-padded 16 x KC tile (coalesced, branchless clamp+select)
  float* sh = tile[wv];
  for (int idx = lane; idx < 16 * KC; idx += 32) {
    int r = idx / KC, c = idx - r * KC;
    int gc = x0 - R + c;
    int cc = iclamp(gc, 0, W - 1);
    float v = pl[(size_t)(y0 + r) * W + cc];
    sh[r * KP + c] = ((unsigned)gc < (unsigned)W && c < K) ? v : 0.f;
  }
  // L2 warm hint for the next tile-row block (global_prefetch_b8 path)
  if (y0 + 64 < H) __builtin_prefetch(&pl[(size_t)(y0 + 64) * W + x0], 0, 1);
  __syncthreads();

  const int mn = lane & 15;
  const int kh = (lane >> 4) << 1;         // 0 or 2 (A/B VGPR layout)
  v8f acc = {0.f, 0.f, 0.f, 0.f, 0.f, 0.f, 0.f, 0.f};
#pragma unroll
  for (int c = 0; c < NCH; ++c) {
    const int kb = (c << 2) + kh;
    v2f a = *(const v2f*)&sh[mn * KP + kb];          // 8B-aligned LDS read
    v2f bv;
    bv.x = gz[16 + kb - mn];
    bv.y = gz[16 + kb + 1 - mn];
    acc = __builtin_amdgcn_wmma_f32_16x16x4_f32(false, a, false, bv, (short)0, acc,
                                                false, false);
  }
  float* op = out + (size_t)b * H * W;
  const int rb8 = (lane >> 4) << 3;
#pragma unroll
  for (int v = 0; v < 8; ++v)
    op[(size_t)(y0 + rb8 + v) * W + (x0 + mn)] = acc[v];
}

// ---------------------------------------------------------------------------
// Vertical blur: O(16x16) = G(16xK) * B(Kx16),
//   G[m,k] = g[k-m], B[k,n] = In[y0-R+k, x0+n] (zero-padded, staged in LDS).
// ---------------------------------------------------------------------------
template <int R>
__global__ void k_vblur_t(const float* __restrict__ in, float* __restrict__ out,
                          int H, int W, float sig) {
  constexpr int K   = 16 + 2 * R;
  constexpr int NCH = (K + 3) / 4;
  constexpr int KC  = NCH * 4;
  __shared__ float gz[64];
  __shared__ __align__(16) float tile[4][32 * 16];

  const int lane = threadIdx.x;
  const int wv   = threadIdx.y;
  const int ft   = wv * 32 + lane;

  build_band<R>(gz, ft, sig);

  const int b  = blockIdx.z;
  const int x0 = blockIdx.x << 4;
  const int y0 = ((blockIdx.y << 2) + wv) << 4;
  const float* pl = in + (size_t)b * H * W;

  // stage zero-padded KC x 16 tile (rows y0-R .. y0+15+R), coalesced
  float* sh = tile[wv];
  for (int idx = lane; idx < KC * 16; idx += 32) {
    int r = idx >> 4, c = idx & 15;
    int gy = y0 - R + r;
    int cy = iclamp(gy, 0, H - 1);
    float v = pl[(size_t)cy * W + x0 + c];
    sh[idx] = ((unsigned)gy < (unsigned)H && r < K) ? v : 0.f;
  }
  if (y0 + 64 < H) __builtin_prefetch(&pl[(size_t)(y0 + 64) * W + x0], 0, 1);
  __syncthreads();

  const int mn = lane & 15;
  const int kh = (lane >> 4) << 1;
  v8f acc = {0.f, 0.f, 0.f, 0.f, 0.f, 0.f, 0.f, 0.f};
#pragma unroll
  for (int c = 0; c < NCH; ++c) {
    const int kb = (c << 2) + kh;
    v2f a;
    a.x = gz[16 + kb - mn];
    a.y = gz[16 + kb + 1 - mn];
    v2f bv;
    bv.x = sh[(kb)     * 16 + mn];
    bv.y = sh[(kb + 1) * 16 + mn];
    acc = __builtin_amdgcn_wmma_f32_16x16x4_f32(false, a, false, bv, (short)0, acc,
                                                false, false);
  }
  float* op = out + (size_t)b * H * W;
  const int rb8 = (lane >> 4) << 3;
#pragma unroll
  for (int v = 0; v < 8; ++v)
    op[(size_t)(y0 + rb8 + v) * W + (x0 + mn)] = acc[v];
}

// ---------------------------------------------------------------------------
// Hessian determinant * sigma^4 (zero-padded 3x3 stencil, branchless)
// ---------------------------------------------------------------------------
__global__ void k_hessdet(const float* __restrict__ in, float* __restrict__ resp,
                          int H, int W, int lvl) {
  int idx = blockIdx.x * blockDim.x + threadIdx.x;
  int total = BATCH * H * W;
  if (idx >= total) return;
  int b = idx / (H * W);
  int p = idx - b * (H * W);
  int y = p / W, x = p - y * W;
  const float* pl = in + (size_t)b * H * W;
  auto ld = [&](int yy, int xx) -> float {
    int cy = iclamp(yy, 0, H - 1), cx2 = iclamp(xx, 0, W - 1);
    float v = pl[(size_t)cy * W + cx2];
    return ((unsigned)yy < (unsigned)H && (unsigned)xx < (unsigned)W) ? v : 0.f;
  };
  float c   = pl[p];
  float dxx = ld(y, x - 1) + ld(y, x + 1) - 2.f * c;
  float dyy = ld(y - 1, x) + ld(y + 1, x) - 2.f * c;
  float dxy = 0.25f * (ld(y - 1, x - 1) - ld(y - 1, x + 1)
                     - ld(y + 1, x - 1) + ld(y + 1, x + 1));
  float s  = d_sig[lvl];
  float s2 = s * s;
  resp[((size_t)b * 3 + lvl) * H * W + p] = (dxx * dyy - dxy * dxy) * (s2 * s2);
}

// ---------------------------------------------------------------------------
// Soft-NMS over 3x3x3 windows, stride 2, softmax(TEMP*patch); emits
// candidates with octave scale folded in plus sortable 64-bit keys.
// ---------------------------------------------------------------------------
__global__ void k_softnms(const float* __restrict__ resp, int H, int W, int oct,
                          int cbase, float* __restrict__ cv, float* __restrict__ cs,
                          float* __restrict__ cx, float* __restrict__ cy,
                          u64* __restrict__ keys) {
  const int Hr = H >> 1, Wr = W >> 1;
  int idx = blockIdx.x * blockDim.x + threadIdx.x;
  int total = BATCH * Hr * Wr;
  if (idx >= total) return;
  int b = idx / (Hr * Wr);
  int p = idx - b * (Hr * Wr);
  int i = p / Wr, j = p - i * Wr;
  const float* rb = resp + (size_t)b * 3 * H * W;

  float vals[27];
  float mx = -3.402823e38f;
  int q = 0;
#pragma unroll
  for (int l = 0; l < 3; ++l)
#pragma unroll
    for (int dy = 0; dy < 3; ++dy)
#pragma unroll
      for (int dx = 0; dx < 3; ++dx) {
        int y = 2 * i - 1 + dy, x = 2 * j - 1 + dx;
        int cy2 = iclamp(y, 0, H - 1), cx2 = iclamp(x, 0, W - 1);
        float v = rb[((size_t)l * H + cy2) * W + cx2];
        v = ((unsigned)y < (unsigned)H && (unsigned)x < (unsigned)W) ? v : 0.f;
        vals[q++] = v;
        mx = fmaxf(mx, TEMP_F * v);
      }
  float S = 0.f, sv = 0.f, sy = 0.f, sx = 0.f, ss = 0.f;
  q = 0;
#pragma unroll
  for (int l = 0; l < 3; ++l)
#pragma unroll
    for (int dy = 0; dy < 3; ++dy)
#pragma unroll
      for (int dx = 0; dx < 3; ++dx) {
        float v  = vals[q++];
        float pr = expf(TEMP_F * v - mx);
        S  += pr;
        sv += pr * v;
        sy += pr * (float)(2 * i - 1 + dy);
        sx += pr * (float)(2 * j - 1 + dx);
        ss += pr * d_sig[l];
      }
  float invS  = 1.f / S;
  float vout  = sv * invS;
  float scale = (float)(1 << oct);
  int ci = cbase + i * Wr + j;
  int g  = b * NCAND + ci;
  cv[g] = vout;
  cs[g] = ss * invS * MR_SIZE * scale;
  cx[g] = sx * invS * scale;
  cy[g] = sy * invS * scale;
  unsigned ub = __float_as_uint(vout);
  unsigned ok = (ub & 0x80000000u) ? ~ub : (ub | 0x80000000u);  // monotone map
  keys[g] = ((u64)ok << 32) | (unsigned)(~ci);                  // ties -> small idx
}

// ---------------------------------------------------------------------------
// In-LDS descending bitonic sort of 4096 u64 keys (256 threads)
// ---------------------------------------------------------------------------
__device__ __forceinline__ void bitonic_sort_desc_4096(u64* sh) {
  for (int k = 2; k <= CHUNK; k <<= 1) {
    for (int j = k >> 1; j > 0; j >>= 1) {
      __syncthreads();
      for (int i = threadIdx.x; i < CHUNK; i += blockDim.x) {
        int l = i ^ j;
        if (l > i) {
          u64 a = sh[i], c = sh[l];
          bool desc = ((i & k) == 0);
          bool sw   = desc ? (a < c) : (a > c);
          if (sw) { sh[i] = c; sh[l] = a; }
        }
      }
    }
  }
  __syncthreads();
}

__global__ void k_topk_reduce(const u64* __restrict__ in, u64* __restrict__ out,
                              int n_in, int stride_in, int stride_out) {
  __shared__ u64 sh[CHUNK];
  const int b = blockIdx.y, ch = blockIdx.x;
  const u64* src = in + (size_t)b * stride_in + (size_t)ch * CHUNK;
  int avail = n_in - ch * CHUNK;
  if (avail > CHUNK) avail = CHUNK;
  for (int t = threadIdx.x; t < CHUNK; t += blockDim.x)
    sh[t] = (t < avail) ? src[t] : 0ULL;
  bitonic_sort_desc_4096(sh);
  u64* dst = out + (size_t)b * stride_out + (size_t)ch * KEEP;
  for (int t = threadIdx.x; t < KEEP; t += blockDim.x)
    dst[t] = sh[t];
}

__global__ void k_topk_final(const u64* __restrict__ in, int n_in,
                             const float* __restrict__ cv, const float* __restrict__ cs,
                             const float* __restrict__ cx, const float* __restrict__ cy,
                             float* __restrict__ out_lafs, float* __restrict__ out_resps) {
  __shared__ u64 sh[CHUNK];
  const int b = blockIdx.x;
  const u64* src = in + (size_t)b * n_in;
  for (int t = threadIdx.x; t < CHUNK; t += blockDim.x)
    sh[t] = (t < n_in) ? src[t] : 0ULL;
  bitonic_sort_desc_4096(sh);
  for (int t = threadIdx.x; t < NUM_FEATS; t += blockDim.x) {
    u64 k = sh[t];
    unsigned ci = ~(unsigned)(k & 0xffffffffu);
    float v = 0.f, s = 0.f, x = 0.f, y = 0.f;
    if (ci < NCAND) {
      int g = b * NCAND + (int)ci;
      v = cv[g]; s = cs[g]; x = cx[g]; y = cy[g];
    }
    float* L = out_lafs + ((size_t)b * NUM_FEATS + t) * 6;
    L[0] = s; L[1] = 0.f; L[2] = x;
    L[3] = 0.f; L[4] = s; L[5] = y;
    out_resps[b * NUM_FEATS + t] = v;
  }
}

// ---------------------------------------------------------------------------
extern "C" void kernel_launch(void* const* d_in, const int* in_sizes, int n_in,
                              void* d_out, int out_size, void* d_ws, size_t ws_size,
                              hipStream_t stream) {
  (void)in_sizes; (void)n_in; (void)out_size; (void)ws_size;
  const float* img = (const float*)d_in[0];

  float* ws = (float*)d_ws;
  size_t off = 0;
  auto alloc = [&](size_t nfloats) -> float* { float* p = ws + off; off += nfloats; return p; };

  float* base1 = alloc((size_t)BATCH * 512 * 512);
  float* base2 = alloc((size_t)BATCH * 256 * 256);
  float* base3 = alloc((size_t)BATCH * 128 * 128);
  float* base4 = alloc((size_t)BATCH * 64 * 64);
  float* bases[5] = {nullptr, base1, base2, base3, base4};
  float* tmpA = alloc((size_t)BATCH * 1024 * 1024);
  float* tmpB = alloc((size_t)BATCH * 1024 * 1024);
  float* resp = alloc((size_t)BATCH * 3 * 1024 * 1024);
  float* cv = alloc((size_t)BATCH * NCAND);
  float* cs = alloc((size_t)BATCH * NCAND);
  float* cx = alloc((size_t)BATCH * NCAND);
  float* cy = alloc((size_t)BATCH * NCAND);
  u64* keys0 = (u64*)(ws + off); off += 2 * (size_t)BATCH * NCAND;
  const int KRED = 176128;  // 86 chunks * 2048 after first reduction round
  u64* kA = (u64*)(ws + off); off += 2 * (size_t)BATCH * KRED;
  u64* kB = (u64*)(ws + off); off += 2 * (size_t)BATCH * KRED;

  const float sg3[3] = {1.6f, 2.0158736798317971f, 2.5398416831491192f};

  const float* cur = img;
  int H = 1024, W = 1024, cbase = 0;
  for (int o = 0; o < 5; ++o) {
    dim3 grd(W / 16, H / 64, BATCH);
    dim3 blk(32, 4);
    for (int lvl = 0; lvl < 3; ++lvl) {
      switch (lvl) {
        case 0:
          k_hblur_t<5><<<grd, blk, 0, stream>>>(cur, tmpA, H, W, sg3[0]);
          k_vblur_t<5><<<grd, blk, 0, stream>>>(tmpA, tmpB, H, W, sg3[0]);
          break;
        case 1:
          k_hblur_t<6><<<grd, blk, 0, stream>>>(cur, tmpA, H, W, sg3[1]);
          k_vblur_t<6><<<grd, blk, 0, stream>>>(tmpA, tmpB, H, W, sg3[1]);
          break;
        default:
          k_hblur_t<8><<<grd, blk, 0, stream>>>(cur, tmpA, H, W, sg3[2]);
          k_vblur_t<8><<<grd, blk, 0, stream>>>(tmpA, tmpB, H, W, sg3[2]);
          break;
      }
      int n = BATCH * H * W;
      k_hessdet<<<(n + 255) / 256, 256, 0, stream>>>(tmpB, resp, H, W, lvl);
    }
    int Hr = H / 2, Wr = W / 2;
    int n = BATCH * Hr * Wr;
    k_softnms<<<(n + 255) / 256, 256, 0, stream>>>(resp, H, W, o, cbase,
                                                   cv, cs, cx, cy, keys0);
    cbase += Hr * Wr;
    if (o < 4) {
      int n2 = BATCH * Hr * Wr;
      k_downsample<<<(n2 + 255) / 256, 256, 0, stream>>>(cur, bases[o + 1], H, W);
      cur = bases[o + 1];
      H = Hr; W = Wr;
    }
  }

  // hierarchical top-k: 349184 -> ... -> 4096, then final sort + gather
  const u64* kin = keys0;
  int n = NCAND, stride = NCAND;
  u64* pong[2] = {kA, kB};
  int pi = 0;
  while (n > CHUNK) {
    int chunks = (n + CHUNK - 1) / CHUNK;
    k_topk_reduce<<<dim3(chunks, BATCH), 256, 0, stream>>>(kin, pong[pi], n, stride,
                                                           chunks * KEEP);
    kin = pong[pi];
    n = chunks * KEEP;
    stride = n;
    pi ^= 1;
  }
  float* out_lafs  = (float*)d_out;
  float* out_resps = out_lafs + (size_t)BATCH * NUM_FEATS * 6;
  k_topk_final<<<BATCH, 256, 0, stream>>>(kin, n, cv, cs, cx, cy, out_lafs, out_resps);
}